// ST_BIFNodeATGF_SS_CUDAOp_34376918237376
// MI455X (gfx1250) — compile-verified
//
#include <hip/hip_runtime.h>

// ST-BIF single-step neuron update (elementwise, memory-bound).
//   H  = V_prev + x
//   s  = +1 if (H >= v_th && T_prev < T_max)
//        -1 if (H < 0    && T_prev > T_min)
//         0 otherwise
//   V  = H - s * v_th
//   T  = T_prev + s
//
// MI455X roofline: 805 MB streamed once / 23.3 TB/s ~= 34.6 us floor.
// Pure bandwidth problem -> 128-bit global loads/stores with non-temporal
// cache hints (single-use data; keep it out of the 192MB L2 / WGP$),
// wave32-sized blocks, branch-free VALU body.

typedef float float4v __attribute__((ext_vector_type(4)));

__global__ __launch_bounds__(256) void st_bif_ss_kernel(
    const float4v* __restrict__ x4,      // x_t
    const float4v* __restrict__ v4,      // V_{t-1}
    const float4v* __restrict__ t4,      // T_{t-1}
    const float*   __restrict__ vth_p,   // v_th scalar
    const float*   __restrict__ tmax_p,  // T_max scalar
    const float*   __restrict__ tmin_p,  // T_min scalar
    float4v*       __restrict__ spike4,  // out: spike
    float4v*       __restrict__ vout4,   // out: V_t
    float4v*       __restrict__ tout4,   // out: T_t
    int n4)                              // number of float4 elements
{
    const int i = blockIdx.x * blockDim.x + threadIdx.x;
    if (i >= n4) return;

    // Uniform scalars: single-element inputs, hoisted out of element math.
    const float vth  = vth_p[0];
    const float tmax = tmax_p[0];
    const float tmin = tmin_p[0];

    // 128-bit non-temporal loads: global_load_b128 th:TH_LOAD_NT.
    // 512B contiguous per wave32 per access; data is touched exactly once.
    const float4v xv = __builtin_nontemporal_load(&x4[i]);
    const float4v vv = __builtin_nontemporal_load(&v4[i]);
    const float4v tv = __builtin_nontemporal_load(&t4[i]);

    float4v s, vn, tn;
#pragma unroll
    for (int k = 0; k < 4; ++k) {
        const float H  = vv[k] + xv[k];
        // Branch-free spike selection (lowers to v_cmp + v_cndmask).
        const float sp = ((H >= vth)  && (tv[k] < tmax)) ? 1.0f : 0.0f;
        const float sn = ((H <  0.0f) && (tv[k] > tmin)) ? 1.0f : 0.0f;
        const float sk = sp - sn;
        s[k]  = sk;
        vn[k] = __builtin_fmaf(-sk, vth, H);  // soft reset: H - s*v_th
        tn[k] = tv[k] + sk;                   // counter update
    }

    // 128-bit non-temporal stores: global_store_b128 th:TH_STORE_NT.
    __builtin_nontemporal_store(s,  &spike4[i]);
    __builtin_nontemporal_store(vn, &vout4[i]);
    __builtin_nontemporal_store(tn, &tout4[i]);
}

extern "C" void kernel_launch(void* const* d_in, const int* in_sizes, int n_in,
                              void* d_out, int out_size, void* d_ws, size_t ws_size,
                              hipStream_t stream) {
    (void)n_in; (void)out_size; (void)d_ws; (void)ws_size;

    const int n  = in_sizes[0];      // B*S*D = 8*4096*1024 = 33,554,432 (divisible by 4)
    const int n4 = n >> 2;

    const float4v* x4 = (const float4v*)d_in[0];
    const float4v* v4 = (const float4v*)d_in[1];
    const float4v* t4 = (const float4v*)d_in[2];
    const float* vth  = (const float*)d_in[3];
    const float* tmax = (const float*)d_in[4];
    const float* tmin = (const float*)d_in[5];

    // d_out = [spike | V_t | T_t], each n floats, in reference return order.
    float* out = (float*)d_out;
    float4v* spike4 = (float4v*)(out);
    float4v* vout4  = (float4v*)(out + (size_t)n);
    float4v* tout4  = (float4v*)(out + 2 * (size_t)n);

    const int block = 256;                      // 8 wave32 waves per block
    const int grid  = (n4 + block - 1) / block; // exact cover: 32,768 blocks

    st_bif_ss_kernel<<<grid, block, 0, stream>>>(
        x4, v4, t4, vth, tmax, tmin, spike4, vout4, tout4, n4);
}